// Inference_4655744549446
// MI455X (gfx1250) — compile-verified
//
#include <hip/hip_runtime.h>
#include <hip/hip_bf16.h>

#define E_DIM 69120
#define N1 60
#define HDIM 32
#define RE_CHUNKS 10
#define RE_EPT 27   // 10 * 27 * 256 == 69120

typedef __attribute__((ext_vector_type(2))) float v2f;
typedef __attribute__((ext_vector_type(8))) float v8f;

__device__ __forceinline__ float relu_(float x) { return x > 0.f ? x : 0.f; }

// ---------------------------------------------------------------------------
// K1: m_bar[e,h] = (1/60) * sum_n relu(d[n,e]*wfv[h] + bfv[h])
// ---------------------------------------------------------------------------
__global__ void __launch_bounds__(256) k_mbar(const float* __restrict__ d,
                                              const float* __restrict__ wfv,
                                              const float* __restrict__ bfv,
                                              float* __restrict__ mbar) {
  const int e = blockIdx.x * 256 + threadIdx.x;
  float w[HDIM], b[HDIM], acc[HDIM];
#pragma unroll
  for (int h = 0; h < HDIM; ++h) { w[h] = wfv[h]; b[h] = bfv[h]; acc[h] = 0.f; }
  for (int n = 0; n < N1; ++n) {
    float x = d[(size_t)n * E_DIM + e];
#pragma unroll
    for (int h = 0; h < HDIM; ++h) acc[h] += relu_(fmaf(x, w[h], b[h]));
  }
  float4* o = (float4*)(mbar + (size_t)e * HDIM);
#pragma unroll
  for (int j = 0; j < 8; ++j) {
    float4 v;
    v.x = acc[4 * j + 0] * (1.f / 60.f);
    v.y = acc[4 * j + 1] * (1.f / 60.f);
    v.z = acc[4 * j + 2] * (1.f / 60.f);
    v.w = acc[4 * j + 3] * (1.f / 60.f);
    o[j] = v;
  }
}

// ---------------------------------------------------------------------------
// K2/K6: chained WMMA GEMMs over E rows:
//   t = relu(in @ Wg + bg)   (16x32 tile per wave, V_WMMA_F32_16X16X4_F32)
//   out = t @ W2             (layout flip through LDS between the two)
// ---------------------------------------------------------------------------
__global__ void __launch_bounds__(256) k_chain(const float* __restrict__ in,
                                               const float* __restrict__ Wg,
                                               const float* __restrict__ bg,
                                               const float* __restrict__ W2,
                                               float* __restrict__ out) {
  __shared__ float lds[8][16 * HDIM];
  const int tid = threadIdx.x;
  const int wave = tid >> 5;
  const int lane = tid & 31;
  const int half = lane >> 4;   // 0: lanes 0-15, 1: lanes 16-31
  const int l16 = lane & 15;
  const int r0 = (blockIdx.x * 8 + wave) * 16;  // base row of this wave's tile

  // B fragments for both weight matrices: B[k,n] 4x16 per chunk.
  // VGPR0 = row K=4c (lanes0-15) / K=4c+2 (lanes16-31); VGPR1 = 4c+1 / 4c+3.
  v2f bw[2][8], b2[2][8];
#pragma unroll
  for (int t = 0; t < 2; ++t)
#pragma unroll
    for (int c = 0; c < 8; ++c) {
      int kb = 4 * c + 2 * half;
      int col = t * 16 + l16;
      bw[t][c].x = Wg[kb * HDIM + col];
      bw[t][c].y = Wg[(kb + 1) * HDIM + col];
      b2[t][c].x = W2[kb * HDIM + col];
      b2[t][c].y = W2[(kb + 1) * HDIM + col];
    }

  // A fragments (16x4 per chunk) straight from global, row-major (E x 32).
  v2f a[8];
  const float* inr = in + (size_t)(r0 + l16) * HDIM;
#pragma unroll
  for (int c = 0; c < 8; ++c) {
    int kb = 4 * c + 2 * half;
    a[c].x = inr[kb];
    a[c].y = inr[kb + 1];
  }

  // First GEMM: acc = in@Wg + bias (bias depends only on N = column).
#pragma unroll
  for (int t = 0; t < 2; ++t) {
    float bv = bg[t * 16 + l16];
    v8f acc;
#pragma unroll
    for (int r = 0; r < 8; ++r) acc[r] = bv;
#pragma unroll
    for (int c = 0; c < 8; ++c)
      acc = __builtin_amdgcn_wmma_f32_16x16x4_f32(false, a[c], false, bw[t][c],
                                                  (short)0, acc, false, false);
    // relu, then C-layout (M = r + 8*half, N = l16) -> row-major LDS tile
#pragma unroll
    for (int r = 0; r < 8; ++r)
      lds[wave][(r + 8 * half) * HDIM + t * 16 + l16] = relu_(acc[r]);
  }
  __syncthreads();

  // Second GEMM: A2 from LDS (v_bar tile), C init 0, store to out.
  v2f a2[8];
  const float* lr = &lds[wave][l16 * HDIM];
#pragma unroll
  for (int c = 0; c < 8; ++c) {
    int kb = 4 * c + 2 * half;
    a2[c].x = lr[kb];
    a2[c].y = lr[kb + 1];
  }
#pragma unroll
  for (int t = 0; t < 2; ++t) {
    v8f acc;
#pragma unroll
    for (int r = 0; r < 8; ++r) acc[r] = 0.f;
#pragma unroll
    for (int c = 0; c < 8; ++c)
      acc = __builtin_amdgcn_wmma_f32_16x16x4_f32(false, a2[c], false, b2[t][c],
                                                  (short)0, acc, false, false);
#pragma unroll
    for (int r = 0; r < 8; ++r)
      out[(size_t)(r0 + r + 8 * half) * HDIM + t * 16 + l16] = acc[r];
  }
}

// ---------------------------------------------------------------------------
// K3/K7: part[chunk,n,h] = sum_{e in chunk} relu(pre[e,h] + d[n,e]*wd[h] + b[h])
// Deterministic two-level reduction (no float atomics -> graph-replay stable).
// ---------------------------------------------------------------------------
__global__ void __launch_bounds__(256) k_reduce_e(const float* __restrict__ pre,
                                                  const float* __restrict__ dmat,
                                                  const float* __restrict__ wd_,
                                                  const float* __restrict__ bias,
                                                  float* __restrict__ part) {
  const int tid = threadIdx.x;
  const int n = blockIdx.y;
  const size_t e0 = (size_t)blockIdx.x * (256 * RE_EPT);
  float w[HDIM], b[HDIM], acc[HDIM];
#pragma unroll
  for (int h = 0; h < HDIM; ++h) { w[h] = wd_[h]; b[h] = bias[h]; acc[h] = 0.f; }

  for (int k = 0; k < RE_EPT; ++k) {
    size_t e = e0 + (size_t)k * 256 + tid;
    float x = dmat[(size_t)n * E_DIM + e];
    const float4* pp = (const float4*)(pre + e * HDIM);
#pragma unroll
    for (int j = 0; j < 8; ++j) {
      float4 p4 = pp[j];
      acc[4 * j + 0] += relu_(p4.x + fmaf(x, w[4 * j + 0], b[4 * j + 0]));
      acc[4 * j + 1] += relu_(p4.y + fmaf(x, w[4 * j + 1], b[4 * j + 1]));
      acc[4 * j + 2] += relu_(p4.z + fmaf(x, w[4 * j + 2], b[4 * j + 2]));
      acc[4 * j + 3] += relu_(p4.w + fmaf(x, w[4 * j + 3], b[4 * j + 3]));
    }
  }
  // wave32 butterfly reduce each h
#pragma unroll
  for (int h = 0; h < HDIM; ++h) {
#pragma unroll
    for (int off = 16; off > 0; off >>= 1) acc[h] += __shfl_xor(acc[h], off, 32);
  }
  __shared__ float wsum[8 * HDIM];
  const int wave = tid >> 5, lane = tid & 31;
  if (lane == 0) {
#pragma unroll
    for (int h = 0; h < HDIM; ++h) wsum[wave * HDIM + h] = acc[h];
  }
  __syncthreads();
  if (tid < HDIM) {
    float s = 0.f;
#pragma unroll
    for (int wv = 0; wv < 8; ++wv) s += wsum[wv * HDIM + tid];
    part[((size_t)blockIdx.x * N1 + n) * HDIM + tid] = s;
  }
}

__global__ void k_sum_part(const float* __restrict__ part, float* __restrict__ out) {
  const int n = blockIdx.x, h = threadIdx.x;
  float s = 0.f;
#pragma unroll
  for (int c = 0; c < RE_CHUNKS; ++c) s += part[((size_t)c * N1 + n) * HDIM + h];
  out[n * HDIM + h] = s;
}

// ---------------------------------------------------------------------------
// K4: label branch + u_n + u_pre (tiny: one block, one thread per n)
// ---------------------------------------------------------------------------
__global__ void k_mid(const int* __restrict__ label,
                      const float* __restrict__ wfv, const float* __restrict__ bfv,
                      const float* __restrict__ Wg1, const float* __restrict__ bg1,
                      const float* __restrict__ Wfu, const float* __restrict__ bfu,
                      const float* __restrict__ Wgu, const float* __restrict__ bgu,
                      const float* __restrict__ Wfvc,
                      const float* __restrict__ alpha_sum,
                      float* __restrict__ u_pre) {
  const int n = threadIdx.x;
  if (n >= N1) return;
  int cnt[3] = {0, 0, 0};
  for (int i = 0; i < N1; ++i) cnt[label[i]]++;

  float cbar[3][HDIM], bpre[3][HDIM];
  for (int c = 0; c < 3; ++c) {
    float mc[HDIM];
#pragma unroll
    for (int h = 0; h < HDIM; ++h)
      mc[h] = ((float)cnt[c] * relu_(wfv[h] + bfv[h]) +
               (float)(N1 - cnt[c]) * relu_(bfv[h])) * (1.f / 60.f);
    for (int h2 = 0; h2 < HDIM; ++h2) {
      float s = bg1[h2];
#pragma unroll
      for (int h = 0; h < HDIM; ++h) s += mc[h] * Wg1[h * HDIM + h2];
      cbar[c][h2] = relu_(s);
    }
    for (int h2 = 0; h2 < HDIM; ++h2) {
      float s = 0.f;
#pragma unroll
      for (int h = 0; h < HDIM; ++h) s += cbar[c][h] * Wfu[h * HDIM + h2];
      bpre[c][h2] = s;
    }
  }
  const int ln = label[n];
  float ab[HDIM];
#pragma unroll
  for (int h = 0; h < HDIM; ++h) {
    float wu_d = Wfu[HDIM * HDIM + h];  // W_fu row 32
    float s = 0.f;
    for (int c = 0; c < 3; ++c)
      s += relu_(bpre[c][h] + ((c == ln) ? wu_d : 0.f) + bfu[h]);
    float beta = s * (1.f / 3.f);
    ab[h] = alpha_sum[n * HDIM + h] * (1.f / (float)E_DIM) + beta;
  }
  float un[HDIM];
  for (int h2 = 0; h2 < HDIM; ++h2) {
    float s = bgu[h2];
#pragma unroll
    for (int h = 0; h < HDIM; ++h) s += ab[h] * Wgu[h * HDIM + h2];
    un[h2] = relu_(s);
  }
  for (int h2 = 0; h2 < HDIM; ++h2) {
    float s = 0.f;
#pragma unroll
    for (int h = 0; h < HDIM; ++h) s += un[h] * Wfvc[h * HDIM + h2];
    u_pre[n * HDIM + h2] = s;
  }
}

// ---------------------------------------------------------------------------
// K5: s2[e,h] = (1/60) sum_n relu(u_pre[n,h] + d[n,e]*wv_d[h] + b[h])
// ---------------------------------------------------------------------------
__global__ void __launch_bounds__(256) k_s2(const float* __restrict__ d,
                                            const float* __restrict__ u_pre,
                                            const float* __restrict__ wv_d,
                                            const float* __restrict__ bias,
                                            float* __restrict__ s2) {
  __shared__ float su[N1 * HDIM];
  for (int i = threadIdx.x; i < N1 * HDIM; i += 256) su[i] = u_pre[i];
  __syncthreads();
  const int e = blockIdx.x * 256 + threadIdx.x;
  float w[HDIM], b[HDIM], acc[HDIM];
#pragma unroll
  for (int h = 0; h < HDIM; ++h) { w[h] = wv_d[h]; b[h] = bias[h]; acc[h] = 0.f; }
  for (int n = 0; n < N1; ++n) {
    float x = d[(size_t)n * E_DIM + e];
#pragma unroll
    for (int h = 0; h < HDIM; ++h)
      acc[h] += relu_(fmaf(x, w[h], su[n * HDIM + h] + b[h]));
  }
  float4* o = (float4*)(s2 + (size_t)e * HDIM);
#pragma unroll
  for (int j = 0; j < 8; ++j) {
    float4 v;
    v.x = acc[4 * j + 0] * (1.f / 60.f);
    v.y = acc[4 * j + 1] * (1.f / 60.f);
    v.z = acc[4 * j + 2] * (1.f / 60.f);
    v.w = acc[4 * j + 3] * (1.f / 60.f);
    o[j] = v;
  }
}

// ---------------------------------------------------------------------------
// K8: out[n,:] = relu((z_sum[n,:]/E) @ W_gz + b_gz)
// ---------------------------------------------------------------------------
__global__ void k_final(const float* __restrict__ z_sum,
                        const float* __restrict__ Wgz,
                        const float* __restrict__ bgz,
                        float* __restrict__ out) {
  const int n = threadIdx.x;
  if (n >= N1) return;
  float z[HDIM];
#pragma unroll
  for (int h = 0; h < HDIM; ++h) z[h] = z_sum[n * HDIM + h] * (1.f / (float)E_DIM);
  for (int h2 = 0; h2 < HDIM; ++h2) {
    float s = bgz[h2];
#pragma unroll
    for (int h = 0; h < HDIM; ++h) s += z[h] * Wgz[h * HDIM + h2];
    out[n * HDIM + h2] = relu_(s);
  }
}

// ---------------------------------------------------------------------------
extern "C" void kernel_launch(void* const* d_in, const int* in_sizes, int n_in,
                              void* d_out, int out_size, void* d_ws, size_t ws_size,
                              hipStream_t stream) {
  (void)in_sizes; (void)n_in; (void)out_size; (void)ws_size;
  const float* sup  = (const float*)d_in[0];
  const int*   lab  = (const int*)d_in[1];
  const float* qry  = (const float*)d_in[2];
  const float* Wfvb = (const float*)d_in[3];
  const float* bfvb = (const float*)d_in[4];
  const float* Wgvb = (const float*)d_in[5];
  const float* bgvb = (const float*)d_in[6];
  const float* Wfu  = (const float*)d_in[7];   // rows 0..31 = Wu_v, row 32 = wu_d
  const float* bfu  = (const float*)d_in[8];
  const float* Wgu  = (const float*)d_in[9];
  const float* bgu  = (const float*)d_in[10];
  const float* Wfvc = (const float*)d_in[11];  // rows 0..31 = Wv_u, row 32 = wv_d
  const float* bfvc = (const float*)d_in[12];
  const float* Wgvc = (const float*)d_in[13];
  const float* bgvc = (const float*)d_in[14];
  const float* Wfz  = (const float*)d_in[15];  // rows 0..31 = Wz_v, row 32 = wz_d
  const float* bfz  = (const float*)d_in[16];
  const float* Wgz  = (const float*)d_in[17];
  const float* bgz  = (const float*)d_in[18];

  float* ws = (float*)d_ws;
  const size_t E32 = (size_t)E_DIM * HDIM;
  float* buf0      = ws;                    // m_bar, later s2
  float* buf1      = ws + E32;              // a_pre, later z_pre
  float* part      = ws + 2 * E32;          // RE_CHUNKS*60*32 partials
  float* alpha_sum = part + (size_t)RE_CHUNKS * N1 * HDIM;
  float* u_pre     = alpha_sum + N1 * HDIM;
  float* z_sum     = u_pre + N1 * HDIM;
  float* outp      = (float*)d_out;

  k_mbar<<<E_DIM / 256, 256, 0, stream>>>(sup, Wfvb, bfvb, buf0);
  k_chain<<<E_DIM / 128, 256, 0, stream>>>(buf0, Wgvb, bgvb, Wfu, buf1);
  k_reduce_e<<<dim3(RE_CHUNKS, N1), 256, 0, stream>>>(buf1, sup, Wfu + HDIM * HDIM, bfu, part);
  k_sum_part<<<N1, HDIM, 0, stream>>>(part, alpha_sum);
  k_mid<<<1, 64, 0, stream>>>(lab, Wfvb, bfvb, Wgvb, bgvb, Wfu, bfu, Wgu, bgu,
                              Wfvc, alpha_sum, u_pre);
  k_s2<<<E_DIM / 256, 256, 0, stream>>>(sup, u_pre, Wfvc + HDIM * HDIM, bfvc, buf0);
  k_chain<<<E_DIM / 128, 256, 0, stream>>>(buf0, Wgvc, bgvc, Wfz, buf1);
  k_reduce_e<<<dim3(RE_CHUNKS, N1), 256, 0, stream>>>(buf1, qry, Wfz + HDIM * HDIM, bfz, part);
  k_sum_part<<<N1, HDIM, 0, stream>>>(part, z_sum);
  k_final<<<1, 64, 0, stream>>>(z_sum, Wgz, bgz, outp);
}